// HiPPO_RNN_58737972740466
// MI455X (gfx1250) — compile-verified
//
#include <hip/hip_runtime.h>
#include <hip/hip_bf16.h>
#include <math.h>
#include <stdint.h>

typedef __attribute__((ext_vector_type(2))) float v2f;
typedef __attribute__((ext_vector_type(8))) float v8f;

namespace {
constexpr int kB = 4, kS = 4096, kIN = 1024, kN = 1024, kOUT = 1024;
constexpr int kM = kB * kS;  // 16384 rows for both GEMMs
constexpr int KC = 32;       // K-chunk staged in LDS (32x64 fp32 = 8KB)
}

// CDNA5 hardware transcendental tanh (TRANS32 group). The ISA requires one
// independent op or V_NOP after a trans op before its result is consumed;
// inline asm hides the trans dependency from the compiler, so embed the v_nop.
__device__ __forceinline__ float tanh_hw(float x) {
  float r;
  asm volatile("v_tanh_f32 %0, %1\n\tv_nop" : "=v"(r) : "v"(x));
  return r;
}

// Issue async global->LDS copies for one 32x64 B chunk (8KB).
// 512 b128 transfers per chunk, 4 per thread. Tracked by ASYNCcnt.
__device__ __forceinline__ void stage_b_chunk(const float* __restrict__ Bm, int N,
                                              int n0, int k0, float* buf, int tid) {
#pragma unroll
  for (int j = 0; j < 4; ++j) {
    const int e    = tid + 128 * j;   // b128 transfer index 0..511
    const int krow = e >> 4;          // 16 transfers per 64-float row
    const int c    = (e & 15) << 2;   // starting column within row
    uint64_t ga = (uint64_t)(uintptr_t)(Bm + (size_t)(k0 + krow) * N + n0 + c);
    uint32_t la = (uint32_t)(uintptr_t)(buf + e * 4);  // LDS byte offset (low 32b)
    asm volatile("global_load_async_to_lds_b128 %0, %1, off"
                 :: "v"(la), "v"(ga) : "memory");
  }
}

// C[M,N] = A[M,K] * B[K,N] (+ bias[N]) via V_WMMA_F32_16X16X4_F32.
// 128 threads = 4 waves; each wave owns 16(M) x 64(N), workgroup covers 64x64.
// B panel double-buffered in LDS via async DMA; A fragments for a whole chunk
// are batch-loaded into registers before the async-wait + barrier so their
// global latency overlaps the staging sync instead of stalling each quad.
__global__ __launch_bounds__(128) void gemm_wmma_f32(
    const float* __restrict__ A, const float* __restrict__ Bm,
    const float* __restrict__ bias, float* __restrict__ C,
    int M, int K, int N) {
  __shared__ __align__(16) float bsh[2][KC * 64];

  const int tid  = threadIdx.x;
  const int lane = tid & 31;
  const int wave = tid >> 5;
  const int lm   = lane & 15;
  const int hi   = lane >> 4;   // 0: lanes 0-15, 1: lanes 16-31
  const int kb   = hi << 1;     // K sub-offset within a quad: 0 or 2
  const int n0   = blockIdx.x * 64;
  const int m0   = blockIdx.y * 64 + wave * 16;

  v8f acc0 = {}, acc1 = {}, acc2 = {}, acc3 = {};
  const float* Arow = A + (size_t)(m0 + lm) * K + kb;

  const int nch = K / KC;
  stage_b_chunk(Bm, N, n0, 0, &bsh[0][0], tid);

  for (int i = 0; i < nch; ++i) {
    const float* cur = &bsh[i & 1][0];

    // Batch-issue all A fragments of this chunk (8 x global_load_b64);
    // they complete while we wait on the async B staging + barrier below.
    const float* Ab = Arow + i * KC;
    v2f a[KC / 4];
#pragma unroll
    for (int q = 0; q < KC / 4; ++q) a[q] = *(const v2f*)(Ab + 4 * q);

    if (i + 1 < nch) {
      stage_b_chunk(Bm, N, n0, (i + 1) * KC, &bsh[(i + 1) & 1][0], tid);
      asm volatile("s_wait_asynccnt 0x4" ::: "memory");  // chunk i done; 4 newer in flight
    } else {
      asm volatile("s_wait_asynccnt 0x0" ::: "memory");
    }
    __syncthreads();  // all waves finished filling buffer i

#pragma unroll
    for (int q = 0; q < KC / 4; ++q) {
      const float* r0 = cur + (4 * q + kb) * 64 + lm;     // B row K = 4q+kb
      const float* r1 = r0 + 64;                          // B row K = 4q+kb+1
      v2f b0 = { r0[0],  r1[0]  };
      v2f b1 = { r0[16], r1[16] };
      v2f b2 = { r0[32], r1[32] };
      v2f b3 = { r0[48], r1[48] };
      acc0 = __builtin_amdgcn_wmma_f32_16x16x4_f32(false, a[q], false, b0, (short)0, acc0, false, false);
      acc1 = __builtin_amdgcn_wmma_f32_16x16x4_f32(false, a[q], false, b1, (short)0, acc1, false, false);
      acc2 = __builtin_amdgcn_wmma_f32_16x16x4_f32(false, a[q], false, b2, (short)0, acc2, false, false);
      acc3 = __builtin_amdgcn_wmma_f32_16x16x4_f32(false, a[q], false, b3, (short)0, acc3, false, false);
    }
    __syncthreads();  // buffer i consumed; safe to overwrite next iteration
  }

  float bv0 = 0.f, bv1 = 0.f, bv2 = 0.f, bv3 = 0.f;
  if (bias) {
    bv0 = bias[n0 + lm];
    bv1 = bias[n0 + lm + 16];
    bv2 = bias[n0 + lm + 32];
    bv3 = bias[n0 + lm + 48];
  }

  float* Crow = C + (size_t)(m0 + hi * 8) * N + n0 + lm;
#pragma unroll
  for (int i = 0; i < 8; ++i) {
    Crow[(size_t)i * N +  0] = acc0[i] + bv0;
    Crow[(size_t)i * N + 16] = acc1[i] + bv1;
    Crow[(size_t)i * N + 32] = acc2[i] + bv2;
    Crow[(size_t)i * N + 48] = acc3[i] + bv3;
  }
}

// Per-(b,n) scalar recurrence: h_t = tanh(h_{t-1}*w[n] + xp[b,t,n] + hippo).
// One thread per (b,n); xp overwritten in place; coalesced across n.
// Unrolled x4 so the 4 independent loads issue ahead of the serial tanh chain;
// per-step latency is one v_fma + one hardware v_tanh_f32.
__global__ __launch_bounds__(256) void rnn_scan(
    float* __restrict__ xp, const float* __restrict__ w,
    const float* __restrict__ u, const float* __restrict__ h0,
    const float* __restrict__ hippo, float* __restrict__ out_state) {
  const int idx = blockIdx.x * blockDim.x + threadIdx.x;  // b*N + n
  const int b = idx >> 10;
  const int n = idx & (kN - 1);

  const float wn = w[n];
  const float ht = hippo[idx] * u[n];  // hippo_state[B,N,1,1] flat == [B*N]
  float h = h0[idx];

  float* p = xp + (size_t)b * kS * kN + n;
  for (int t = 0; t < kS; t += 4) {
    float v0 = p[0 * kN] + ht;   // independent of h: loads + adds overlap the chain
    float v1 = p[1 * kN] + ht;
    float v2 = p[2 * kN] + ht;
    float v3 = p[3 * kN] + ht;
    h = tanh_hw(fmaf(h, wn, v0));  p[0 * kN] = h;
    h = tanh_hw(fmaf(h, wn, v1));  p[1 * kN] = h;
    h = tanh_hw(fmaf(h, wn, v2));  p[2 * kN] = h;
    h = tanh_hw(fmaf(h, wn, v3));  p[3 * kN] = h;
    p += 4 * kN;
  }
  out_state[idx] = h;
}

extern "C" void kernel_launch(void* const* d_in, const int* in_sizes, int n_in,
                              void* d_out, int out_size, void* d_ws, size_t ws_size,
                              hipStream_t stream) {
  const float* x     = (const float*)d_in[0];
  const float* Wi    = (const float*)d_in[1];
  const float* bi    = (const float*)d_in[2];
  const float* w     = (const float*)d_in[3];
  const float* u     = (const float*)d_in[4];
  const float* Wo    = (const float*)d_in[5];
  const float* h0    = (const float*)d_in[6];
  const float* hippo = (const float*)d_in[7];

  float* out       = (float*)d_out;
  float* out_state = out + (size_t)kM * kOUT;  // [B,N] tail of d_out
  float* xp        = (float*)d_ws;             // kM*kN floats (64 MB)

  dim3 blk(128);

  // Stage 1: xp = x*Wi + bi
  gemm_wmma_f32<<<dim3(kN / 64, kM / 64), blk, 0, stream>>>(x, Wi, bi, xp, kM, kIN, kN);

  // Stage 2: sequential scan over t, in place; final h -> out_state
  rnn_scan<<<dim3((kB * kN) / 256), dim3(256), 0, stream>>>(xp, w, u, h0, hippo, out_state);

  // Stage 3: out = states*Wo
  gemm_wmma_f32<<<dim3(kOUT / 64, kM / 64), blk, 0, stream>>>(xp, Wo, nullptr, out, kM, kN, kOUT);
}